// RNNSeq_12008728560188
// MI455X (gfx1250) — compile-verified
//
#include <hip/hip_runtime.h>
#include <hip/hip_bf16.h>

// ---------------------------------------------------------------------------
// Problem constants (match reference)
// ---------------------------------------------------------------------------
constexpr int Bn = 2048;
constexpr int Tn = 2048;
constexpr int ROWS = Bn * Tn;            // 4,194,304 (b,t) rows
constexpr int TILES = ROWS / 16;         // 262,144 WMMA row-tiles (exact)
constexpr int TILES_PER_WAVE = 8;

typedef __attribute__((ext_vector_type(2))) float v2f;
typedef __attribute__((ext_vector_type(8))) float v8f;

__device__ __forceinline__ float sigmoidf_(float x) {
    return 1.0f / (1.0f + __expf(-x));
}

// ---------------------------------------------------------------------------
// Kernel 1: xi = emb[x] @ Wih^T + bih for BOTH directions, via
// V_WMMA_F32_16X16X4_F32 (M=16 rows of (b,t), K=4=E, N=16 (12 gates + pad)).
// A layout (32-bit 16x4): lanes 0-15 hold K=0,1 in v[0],v[1]; lanes 16-31 K=2,3.
// B layout (4x16):        lanes 0-15 hold K=0,1; lanes 16-31 hold K=2,3; N=lane%16.
// D layout (16x16 f32):   VGPR r, lane n(<16) -> (M=r, N=n); lane n+16 -> (M=r+8, N=n).
// ---------------------------------------------------------------------------
__global__ void xi_wmma_kernel(const int* __restrict__ x,
                               const float* __restrict__ emb,
                               const float* __restrict__ Wih_f,
                               const float* __restrict__ bih_f,
                               const float* __restrict__ Wih_b,
                               const float* __restrict__ bih_b,
                               float* __restrict__ xi_f,
                               float* __restrict__ xi_b) {
    const int lane  = threadIdx.x & 31;
    const int wave  = (blockIdx.x * blockDim.x + threadIdx.x) >> 5;
    const int n     = lane & 15;   // column (gate index, 12 valid)
    const int khalf = lane >> 4;   // 0: K=0,1 ; 1: K=2,3

    // B operands (loop invariant): B[k][n] = Wih[n][k], zero-padded for n>=12.
    v2f bf; bf[0] = 0.f; bf[1] = 0.f;
    v2f bb; bb[0] = 0.f; bb[1] = 0.f;
    v8f cf = {}; v8f cb = {};
    if (n < 12) {
        bf[0] = Wih_f[n * 4 + 2 * khalf + 0];
        bf[1] = Wih_f[n * 4 + 2 * khalf + 1];
        bb[0] = Wih_b[n * 4 + 2 * khalf + 0];
        bb[1] = Wih_b[n * 4 + 2 * khalf + 1];
        const float bif = bih_f[n];
        const float bib = bih_b[n];
#pragma unroll
        for (int i = 0; i < 8; ++i) { cf[i] = bif; cb[i] = bib; }
    }

    const int tile0 = wave * TILES_PER_WAVE;
#pragma unroll 1
    for (int tt = 0; tt < TILES_PER_WAVE; ++tt) {
        const int m0  = (tile0 + tt) * 16;
        const int row = m0 + n;                       // two lanes per row (K split)
        const int idx = x[row];
        const float4 e = ((const float4*)emb)[idx];   // embedding row, E=4
        v2f a;
        a[0] = khalf ? e.z : e.x;
        a[1] = khalf ? e.w : e.y;

        v8f df = __builtin_amdgcn_wmma_f32_16x16x4_f32(
            false, a, false, bf, (short)0, cf, false, false);
        v8f db = __builtin_amdgcn_wmma_f32_16x16x4_f32(
            false, a, false, bb, (short)0, cb, false, false);

        if (n < 12) {
#pragma unroll
            for (int vr = 0; vr < 8; ++vr) {
                const size_t r = (size_t)(m0 + vr + 8 * khalf);
                xi_f[r * 12 + n] = df[vr];
                xi_b[r * 12 + n] = db[vr];
            }
        }
    }
}

// ---------------------------------------------------------------------------
// Kernel 2: forward + backward GRU scans, run concurrently.
// tid in [0,2B): dir = tid/B (0=fwd, 1=bwd), b = tid%B. One serial chain per
// thread; Whh/bhh held in registers (uniform loads -> SGPRs).
// ---------------------------------------------------------------------------
__global__ void gru_scan_kernel(const float* __restrict__ xi_f,
                                const float* __restrict__ xi_b,
                                const float* __restrict__ Whh_f,
                                const float* __restrict__ bhh_f,
                                const float* __restrict__ Whh_b,
                                const float* __restrict__ bhh_b,
                                float* __restrict__ hf,
                                float* __restrict__ hb) {
    const int tid = blockIdx.x * blockDim.x + threadIdx.x;
    if (tid >= 2 * Bn) return;
    const int dir = tid >> 11;          // 2048 rows per direction
    const int b   = tid & (Bn - 1);

    const float* __restrict__ xi   = dir ? xi_b  : xi_f;
    const float* __restrict__ WhhP = dir ? Whh_b : Whh_f;
    const float* __restrict__ bhhP = dir ? bhh_b : bhh_f;
    float* __restrict__ hout       = dir ? hb    : hf;

    float W[48];
#pragma unroll
    for (int i = 0; i < 48; ++i) W[i] = WhhP[i];
    float bias[12];
#pragma unroll
    for (int i = 0; i < 12; ++i) bias[i] = bhhP[i];

    float h[4] = {0.f, 0.f, 0.f, 0.f};
#pragma unroll 1
    for (int s = 0; s < Tn; ++s) {
        const int t = dir ? (Tn - 1 - s) : s;
        const size_t base = (size_t)b * Tn + t;
        const float* xr = xi + base * 12;           // 48B, 16B-aligned
        const float4 q0 = ((const float4*)xr)[0];
        const float4 q1 = ((const float4*)xr)[1];
        const float4 q2 = ((const float4*)xr)[2];
        const float xg[12] = {q0.x, q0.y, q0.z, q0.w,
                              q1.x, q1.y, q1.z, q1.w,
                              q2.x, q2.y, q2.z, q2.w};
        float g[12];
#pragma unroll
        for (int gi = 0; gi < 12; ++gi) {
            g[gi] = bias[gi] + W[gi * 4 + 0] * h[0] + W[gi * 4 + 1] * h[1]
                             + W[gi * 4 + 2] * h[2] + W[gi * 4 + 3] * h[3];
        }
        float hn[4];
#pragma unroll
        for (int i = 0; i < 4; ++i) {
            const float r = sigmoidf_(xg[i]     + g[i]);
            const float z = sigmoidf_(xg[4 + i] + g[4 + i]);
            const float nv = tanhf(xg[8 + i] + r * g[8 + i]);
            hn[i] = (1.f - z) * nv + z * h[i];
        }
#pragma unroll
        for (int i = 0; i < 4; ++i) h[i] = hn[i];
        float4 o; o.x = h[0]; o.y = h[1]; o.z = h[2]; o.w = h[3];
        ((float4*)hout)[base] = o;
    }
}

// ---------------------------------------------------------------------------
// Kernel 3: output GRU (input 8 = [hf, hb], hidden 1) + final sigmoid.
// ---------------------------------------------------------------------------
__global__ void out_scan_kernel(const float* __restrict__ hf,
                                const float* __restrict__ hb,
                                const float* __restrict__ Wih_o,
                                const float* __restrict__ Whh_o,
                                const float* __restrict__ bih_o,
                                const float* __restrict__ bhh_o,
                                float* __restrict__ out) {
    const int b = blockIdx.x * blockDim.x + threadIdx.x;
    if (b >= Bn) return;

    float Wi[24];
#pragma unroll
    for (int i = 0; i < 24; ++i) Wi[i] = Wih_o[i];
    float Wh[3], bi_[3], bh[3];
#pragma unroll
    for (int i = 0; i < 3; ++i) { Wh[i] = Whh_o[i]; bi_[i] = bih_o[i]; bh[i] = bhh_o[i]; }

    float ho = 0.f;
#pragma unroll 1
    for (int t = 0; t < Tn; ++t) {
        const size_t base = (size_t)b * Tn + t;
        const float4 f4 = ((const float4*)hf)[base];
        const float4 b4 = ((const float4*)hb)[base];
        const float bi8[8] = {f4.x, f4.y, f4.z, f4.w, b4.x, b4.y, b4.z, b4.w};
        float xo[3];
#pragma unroll
        for (int g = 0; g < 3; ++g) {
            float s = bi_[g];
#pragma unroll
            for (int k = 0; k < 8; ++k) s += Wi[g * 8 + k] * bi8[k];
            xo[g] = s;
        }
        const float r  = sigmoidf_(xo[0] + Wh[0] * ho + bh[0]);
        const float z  = sigmoidf_(xo[1] + Wh[1] * ho + bh[1]);
        const float nv = tanhf(xo[2] + r * (Wh[2] * ho + bh[2]));
        ho = (1.f - z) * nv + z * ho;
        out[base] = sigmoidf_(ho);
    }
}

// ---------------------------------------------------------------------------
extern "C" void kernel_launch(void* const* d_in, const int* in_sizes, int n_in,
                              void* d_out, int out_size, void* d_ws, size_t ws_size,
                              hipStream_t stream) {
    (void)in_sizes; (void)n_in; (void)out_size; (void)ws_size;

    const int*   x     = (const int*)  d_in[0];
    const float* emb   = (const float*)d_in[1];
    const float* Wih_f = (const float*)d_in[2];
    // d_in[3] = Whh_f
    const float* Whh_f = (const float*)d_in[3];
    const float* bih_f = (const float*)d_in[4];
    const float* bhh_f = (const float*)d_in[5];
    const float* Wih_b = (const float*)d_in[6];
    const float* Whh_b = (const float*)d_in[7];
    const float* bih_b = (const float*)d_in[8];
    const float* bhh_b = (const float*)d_in[9];
    const float* Wih_o = (const float*)d_in[10];
    const float* Whh_o = (const float*)d_in[11];
    const float* bih_o = (const float*)d_in[12];
    const float* bhh_o = (const float*)d_in[13];
    float* out = (float*)d_out;

    // Workspace carve-up (all regions fully overwritten before read):
    //   xi_f: ROWS*12 f32 (192MB), xi_b: ROWS*12 f32 (192MB),
    //   hf:   ROWS*4  f32 ( 64MB), hb:   ROWS*4  f32 ( 64MB)
    float* xi_f = (float*)d_ws;
    float* xi_b = xi_f + (size_t)ROWS * 12;
    float* hf   = xi_b + (size_t)ROWS * 12;
    float* hb   = hf   + (size_t)ROWS * 4;

    // K1: 262,144 tiles / 8 tiles-per-wave = 32,768 waves; 8 waves/block.
    const int waves   = TILES / TILES_PER_WAVE;
    const int blocks1 = waves / 8;            // 4096 blocks of 256 threads
    xi_wmma_kernel<<<dim3(blocks1), dim3(256), 0, stream>>>(
        x, emb, Wih_f, bih_f, Wih_b, bih_b, xi_f, xi_b);

    // K2: 4096 serial scan threads (fwd+bwd), 1 wave per block to spread WGPs.
    gru_scan_kernel<<<dim3(2 * Bn / 32), dim3(32), 0, stream>>>(
        xi_f, xi_b, Whh_f, bhh_f, Whh_b, bhh_b, hf, hb);

    // K3: 2048 serial output-GRU threads.
    out_scan_kernel<<<dim3(Bn / 32), dim3(32), 0, stream>>>(
        hf, hb, Wih_o, Whh_o, bih_o, bhh_o, out);
}